// Embedding_27676769255484
// MI455X (gfx1250) — compile-verified
//
#include <hip/hip_runtime.h>
#include <cstdint>
#include <cstddef>

#ifndef __has_builtin
#define __has_builtin(x) 0
#endif

// Problem shape (fixed by the reference)
#define SEQ       8192
#define DIM       512
#define WAVES     8                       // waves per block (wave32 -> 256 threads)
#define BLOCKS    512
#define ROWSTRIDE (BLOCKS * WAVES)        // 4096 rows per sweep
#define NROWS     (SEQ / ROWSTRIDE)       // 2 rows per wave

// ---- CDNA5 feature detection -------------------------------------------------
#if __has_builtin(__builtin_amdgcn_global_load_async_to_lds_b128)
#define USE_ASYNC 1
#else
#define USE_ASYNC 0
#endif

#if __has_builtin(__builtin_amdgcn_s_wait_asynccnt)
#define WAIT_ASYNC(N) __builtin_amdgcn_s_wait_asynccnt(N)
#else
#define WAIT_ASYNC(N) asm volatile("s_wait_asynccnt %0" ::"i"(N) : "memory")
#endif

// Native vector types (clang ext vectors, usable with nontemporal builtins)
typedef int   v4i __attribute__((ext_vector_type(4)));
typedef float v4f __attribute__((ext_vector_type(4)));
typedef __attribute__((address_space(1))) v4i* gas_v4i;
typedef __attribute__((address_space(3))) v4i* las_v4i;

// log2(10000), 1/(2*pi)
#define LOG2_10000 13.287712379549449f
#define INV_2PI    0.15915494309189535f

// Issue one 2KB table row (per-wave) as 4x async B128 copies into LDS.
// Each lane moves 16B per instruction: 32 lanes x 16B = 512B per issue.
__device__ __forceinline__ void issue_row_async(const float* gsrc, float* ldst, int lane) {
#if USE_ASYNC
#pragma unroll
  for (int k = 0; k < 4; ++k) {
    const int j = lane + 32 * k;           // float4 slot within the row
    __builtin_amdgcn_global_load_async_to_lds_b128(
        (gas_v4i)(gsrc + 4 * j),
        (las_v4i)(ldst + 4 * j),
        /*imm offset*/ 0, /*cpol*/ 0);
  }
#else
  (void)gsrc; (void)ldst; (void)lane;
#endif
}

// PE(pos, col): angle = pos / 10000^(2*col/DIM); even col -> sin, odd -> cos.
// invf[] is pre-scaled by 1/(2pi): hw v_sin/v_cos take revolutions; v_fract
// range-reduces the (large) argument into one period.
__device__ __forceinline__ float pe_sin(float pos, float invf) {
  return __builtin_amdgcn_sinf(__builtin_amdgcn_fractf(pos * invf));
}
__device__ __forceinline__ float pe_cos(float pos, float invf) {
  return __builtin_amdgcn_cosf(__builtin_amdgcn_fractf(pos * invf));
}

// Streaming (write-once) output: non-temporal B128 store keeps the embedding
// table resident in the 192MB L2 across graph replays.
__device__ __forceinline__ void store_nt(float* dst, v4f v) {
  __builtin_nontemporal_store(v, (v4f*)dst);      // global_store_b128 th:NT
}

__global__ __launch_bounds__(WAVES * 32)
void embed_pe_kernel(const int* __restrict__ x,
                     const float* __restrict__ table,
                     float* __restrict__ out) {
  const int lane = threadIdx.x & 31;
  const int wave = threadIdx.x >> 5;
  const int r0   = blockIdx.x * WAVES + wave;   // 0..4095

  // Per-lane column frequencies: lane handles float4 slots j = lane + 32k,
  // i.e. columns 4j..4j+3. Computed once per wave (v_exp_f32), reused per row.
  float invf[16];
#pragma unroll
  for (int k = 0; k < 4; ++k) {
#pragma unroll
    for (int e = 0; e < 4; ++e) {
      const int col = 4 * (lane + 32 * k) + e;
      invf[4 * k + e] =
          __builtin_amdgcn_exp2f(-(float)col * (2.0f / (float)DIM) * LOG2_10000) * INV_2PI;
    }
  }

#if USE_ASYNC
  // Double-buffered async gather: stream row n+1 into LDS while adding PE to row n.
  __shared__ float smem[WAVES][2][DIM];         // 32 KB / block

  issue_row_async(table + (size_t)x[r0] * DIM, &smem[wave][0][0], lane);

#pragma unroll
  for (int n = 0; n < NROWS; ++n) {
    const int r = r0 + n * ROWSTRIDE;
    if (n + 1 < NROWS) {
      const int rn = r + ROWSTRIDE;
      issue_row_async(table + (size_t)x[rn] * DIM, &smem[wave][(n + 1) & 1][0], lane);
      WAIT_ASYNC(4);                            // row n's 4 copies done (in-order)
    } else {
      WAIT_ASYNC(0);
    }
    const float pos = (float)r;
    const float* buf = &smem[wave][n & 1][0];
    float* dst = out + (size_t)r * DIM;
#pragma unroll
    for (int k = 0; k < 4; ++k) {
      const int j = lane + 32 * k;
      v4f v = *(const v4f*)(buf + 4 * j);       // ds_load_b128
      v.x += pe_sin(pos, invf[4 * k + 0]);
      v.y += pe_cos(pos, invf[4 * k + 1]);
      v.z += pe_sin(pos, invf[4 * k + 2]);
      v.w += pe_cos(pos, invf[4 * k + 3]);
      store_nt(dst + 4 * j, v);                 // global_store_b128 th:NT
    }
  }
#else
  // Fallback: direct B128 loads, prefetch the next gathered row (global_prefetch).
#pragma unroll
  for (int n = 0; n < NROWS; ++n) {
    const int r = r0 + n * ROWSTRIDE;
    const float* src = table + (size_t)x[r] * DIM;
    if (n + 1 < NROWS) {
      const float* nsrc = table + (size_t)x[r + ROWSTRIDE] * DIM;
      __builtin_prefetch(nsrc + lane * 16, 0, 0); // one 64B line per lane = 2KB row
    }
    const float pos = (float)r;
    float* dst = out + (size_t)r * DIM;
#pragma unroll
    for (int k = 0; k < 4; ++k) {
      const int j = lane + 32 * k;
      v4f v = *(const v4f*)(src + 4 * j);       // global_load_b128
      v.x += pe_sin(pos, invf[4 * k + 0]);
      v.y += pe_cos(pos, invf[4 * k + 1]);
      v.z += pe_sin(pos, invf[4 * k + 2]);
      v.w += pe_cos(pos, invf[4 * k + 3]);
      store_nt(dst + 4 * j, v);                 // global_store_b128 th:NT
    }
  }
#endif
}

extern "C" void kernel_launch(void* const* d_in, const int* in_sizes, int n_in,
                              void* d_out, int out_size, void* d_ws, size_t ws_size,
                              hipStream_t stream) {
  (void)in_sizes; (void)n_in; (void)out_size; (void)d_ws; (void)ws_size;
  const int*   x     = (const int*)d_in[0];    // [SEQ] token ids
  const float* table = (const float*)d_in[1];  // [VOCAB, DIM] fp32
  float*       out   = (float*)d_out;          // [SEQ, DIM] fp32
  embed_pe_kernel<<<BLOCKS, WAVES * 32, 0, stream>>>(x, table, out);
}